// LocSE_54528904790898
// MI455X (gfx1250) — compile-verified
//
#include <hip/hip_runtime.h>
#include <math.h>

typedef __attribute__((ext_vector_type(2))) float v2f;
typedef __attribute__((ext_vector_type(8))) float v8f;

// Problem constants (from reference setup_inputs)
#define BSZ   8
#define NPTS  32768
#define KNB   16
#define NK    (NPTS * KNB)        // 524288 = 1<<19
#define LTOT  (BSZ * NK)          // 4194304
#define CIN   10
#define COUT  32
#define KPAD  12                  // CIN padded to 3 WMMA K-steps of 4
#define EPSV  1e-5f
#define SLOPE 0.01f

// workspace float layout
#define WS_STATS 0      // 65 floats: S1[10], S2tri[55]
#define WS_WP    80     // 32*12 = 384 floats (BN-folded, K-padded weights)
#define WS_BP    (80 + 384) // 32 floats (BN-folded bias)

// ---------------------------------------------------------------- init stats
__global__ void locse_init(float* __restrict__ stats) {
    int i = threadIdx.x;
    if (i < 65) stats[i] = 0.0f;
}

// ------------------------------------------------------- wave32 butterfly sum
__device__ __forceinline__ float wave_red(float v) {
#pragma unroll
    for (int off = 16; off >= 1; off >>= 1)
        v += __shfl_xor(v, off, 32);
    return v;
}

// ----------------------------------------------- pass 1: feature 2nd moments
// S1[c]   = sum_p f_c(p)
// S2[c,d] = sum_p f_c(p) f_d(p)   (upper triangle, 55 entries)
__global__ void __launch_bounds__(256)
locse_stats(const float* __restrict__ xyz, const int* __restrict__ nidx,
            float* __restrict__ stats) {
    const int tid = blockIdx.x * blockDim.x + threadIdx.x;
    const int nth = gridDim.x * blockDim.x;

    float s1[10];
    float s2[55];
#pragma unroll
    for (int i = 0; i < 10; i++) s1[i] = 0.0f;
#pragma unroll
    for (int i = 0; i < 55; i++) s2[i] = 0.0f;

    for (int p = tid; p < LTOT; p += nth) {
        const int b = p >> 19;          // / NK
        const int l = p & (NK - 1);
        const int n = l >> 4;           // / KNB
        const float* xb = xyz + (size_t)b * 3 * NPTS;
        const int m = nidx[(size_t)b * NK + l];
        const float c0 = xb[n], c1 = xb[NPTS + n], c2 = xb[2 * NPTS + n];
        const float q0 = xb[m], q1 = xb[NPTS + m], q2 = xb[2 * NPTS + m];
        const float r0 = c0 - q0, r1 = c1 - q1, r2 = c2 - q2;
        const float dist = sqrtf(r0 * r0 + r1 * r1 + r2 * r2);
        const float f[10] = {dist, r0, r1, r2, c0, c1, c2, q0, q1, q2};
        int t = 0;
#pragma unroll
        for (int c = 0; c < 10; c++) {
            s1[c] += f[c];
#pragma unroll
            for (int d = c; d < 10; d++)
                s2[t++] += f[c] * f[d];
        }
    }

#pragma unroll
    for (int i = 0; i < 10; i++) s1[i] = wave_red(s1[i]);
#pragma unroll
    for (int i = 0; i < 55; i++) s2[i] = wave_red(s2[i]);

    if ((threadIdx.x & 31) == 0) {
#pragma unroll
        for (int i = 0; i < 10; i++) atomicAdd(&stats[i], s1[i]);
#pragma unroll
        for (int i = 0; i < 55; i++) atomicAdd(&stats[10 + i], s2[i]);
    }
}

// ------------------------- pass 1b: fold BatchNorm into conv weights (1 blk)
// mean_o = (W_o . S1)/L + b_o
// E[h^2] = (W_o^T S2 W_o)/L + 2 b_o (W_o.S1)/L + b_o^2 ; var = E[h^2]-mean^2
// Wp = gamma*rsqrt(var+eps)*W ; bp = gamma*rsqrt(var+eps)*(b-mean)+beta
__global__ void locse_prep(const float* __restrict__ W, const float* __restrict__ bias,
                           const float* __restrict__ gamma, const float* __restrict__ beta,
                           const float* __restrict__ stats,
                           float* __restrict__ Wp, float* __restrict__ bp) {
    const int o = threadIdx.x;
    if (o >= COUT) return;
    const float invL = 1.0f / (float)LTOT;
    float w[10];
#pragma unroll
    for (int c = 0; c < 10; c++) w[c] = W[o * CIN + c];

    float s1w = 0.0f;
#pragma unroll
    for (int c = 0; c < 10; c++) s1w += w[c] * stats[c];

    const float bo = bias[o];
    const float mean = s1w * invL + bo;

    float q = 0.0f;
    int t = 10;
    for (int c = 0; c < 10; c++)
        for (int d = c; d < 10; d++) {
            const float v = stats[t++];
            q += w[c] * w[d] * v * ((d == c) ? 1.0f : 2.0f);
        }
    const float Eh2 = q * invL + 2.0f * bo * s1w * invL + bo * bo;
    const float var = Eh2 - mean * mean;
    const float scale = gamma[o] * rsqrtf(var + EPSV);

#pragma unroll
    for (int c = 0; c < KPAD; c++)
        Wp[o * KPAD + c] = (c < CIN) ? w[c] * scale : 0.0f;
    bp[o] = (bo - mean) * scale + beta[o];
}

// --------------------------------- pass 2: fused gather + WMMA conv + store
// Per wave tile: 16 points (same b,n; k=0..15). h[32x16] = Wp[32x12] x f[12x16]
// via V_WMMA_F32_16X16X4_F32: 2 M-tiles x 3 K-steps.
// A (16x4 f32, 2 VGPRs): lanes0-15 M-rows with K={4s,4s+1}; lanes16-31 K={4s+2,4s+3}
// B (4x16 f32, 2 VGPRs): lanes0-15 rows K=4s (x) / 4s+1 (y); lanes16-31 K=4s+2 / 4s+3
// C/D (16x16 f32, 8 VGPRs): VGPR r -> row M=r (lanes<16) / M=r+8 (lanes>=16)
__global__ void __launch_bounds__(256)
locse_main(const float* __restrict__ xyz, const int* __restrict__ nidx,
           const float* __restrict__ Wp, const float* __restrict__ bp,
           float* __restrict__ out) {
    const int lane = threadIdx.x & 31;
    const int lo   = lane & 15;
    const int half = lane >> 4;          // 0 | 1
    const bool hi  = (half != 0);
    const int wave   = (blockIdx.x * blockDim.x + threadIdx.x) >> 5;
    const int nwaves = (gridDim.x * blockDim.x) >> 5;

    // Preload A-matrix tiles of folded weights (constant across loop)
    v2f A[2][3];
#pragma unroll
    for (int mt = 0; mt < 2; mt++)
#pragma unroll
        for (int s = 0; s < 3; s++) {
            const int row = mt * 16 + lo;
            const int kb = 4 * s + half * 2;
            A[mt][s].x = Wp[row * KPAD + kb];
            A[mt][s].y = Wp[row * KPAD + kb + 1];
        }
    // Per-lane bias for each D VGPR row
    float biasr[2][8];
#pragma unroll
    for (int mt = 0; mt < 2; mt++)
#pragma unroll
        for (int r = 0; r < 8; r++)
            biasr[mt][r] = bp[mt * 16 + r + half * 8];

    const int ntiles = LTOT / 16;        // 262144
    for (int t = wave; t < ntiles; t += nwaves) {
        const int p0 = t * 16;
        const int b  = p0 >> 19;
        const int l0 = p0 & (NK - 1);
        const int n  = l0 >> 4;          // all 16 points of the tile share (b,n)
        const float* xb = xyz + (size_t)b * 3 * NPTS;

        const int m = nidx[(size_t)b * NK + l0 + lo];   // coalesced 16 ints
        const float c0 = xb[n],        c1 = xb[NPTS + n], c2 = xb[2 * NPTS + n];
        const float q0 = xb[m],        q1 = xb[NPTS + m], q2 = xb[2 * NPTS + m];
        const float r0 = c0 - q0, r1 = c1 - q1, r2 = c2 - q2;
        const float f0 = sqrtf(r0 * r0 + r1 * r1 + r2 * r2);

        // Build B-matrix registers in the wmma lane layout
        v2f B0, B1, B2;
        B0.x = hi ? r1 : f0;  B0.y = hi ? r2 : r0;    // K 0..3 : dist,r0,r1,r2
        B1.x = hi ? c2 : c0;  B1.y = hi ? q0 : c1;    // K 4..7 : c0,c1,c2,q0
        B2.x = hi ? 0.f : q1; B2.y = hi ? 0.f : q2;   // K 8..11: q1,q2,0,0

        v8f acc0 = {0.f, 0.f, 0.f, 0.f, 0.f, 0.f, 0.f, 0.f};
        v8f acc1 = {0.f, 0.f, 0.f, 0.f, 0.f, 0.f, 0.f, 0.f};
        acc0 = __builtin_amdgcn_wmma_f32_16x16x4_f32(false, A[0][0], false, B0, (short)0, acc0, false, false);
        acc0 = __builtin_amdgcn_wmma_f32_16x16x4_f32(false, A[0][1], false, B1, (short)0, acc0, false, false);
        acc0 = __builtin_amdgcn_wmma_f32_16x16x4_f32(false, A[0][2], false, B2, (short)0, acc0, false, false);
        acc1 = __builtin_amdgcn_wmma_f32_16x16x4_f32(false, A[1][0], false, B0, (short)0, acc1, false, false);
        acc1 = __builtin_amdgcn_wmma_f32_16x16x4_f32(false, A[1][1], false, B1, (short)0, acc1, false, false);
        acc1 = __builtin_amdgcn_wmma_f32_16x16x4_f32(false, A[1][2], false, B2, (short)0, acc1, false, false);

        // Bias + LeakyReLU + store: out[b, o, l0+lo], two 64B segments per store
        const size_t outb = (size_t)b * COUT * NK + (size_t)(l0 + lo);
#pragma unroll
        for (int r = 0; r < 8; r++) {
            const int o0 = r + half * 8;
            float v = acc0[r] + biasr[0][r];
            v = (v >= 0.f) ? v : SLOPE * v;
            out[outb + (size_t)o0 * NK] = v;
            const int o1 = 16 + r + half * 8;
            float w = acc1[r] + biasr[1][r];
            w = (w >= 0.f) ? w : SLOPE * w;
            out[outb + (size_t)o1 * NK] = w;
        }
    }
}

// --------------------------------------------------------------------- launch
extern "C" void kernel_launch(void* const* d_in, const int* in_sizes, int n_in,
                              void* d_out, int out_size, void* d_ws, size_t ws_size,
                              hipStream_t stream) {
    const float* xyz   = (const float*)d_in[0];
    const int*   nidx  = (const int*)d_in[1];
    const float* W     = (const float*)d_in[2];
    const float* bias  = (const float*)d_in[3];
    const float* gamma = (const float*)d_in[4];
    const float* beta  = (const float*)d_in[5];
    float* out = (float*)d_out;

    float* ws    = (float*)d_ws;
    float* stats = ws + WS_STATS;
    float* Wp    = ws + WS_WP;
    float* bp    = ws + WS_BP;

    locse_init <<<1, 128, 0, stream>>>(stats);
    locse_stats<<<512, 256, 0, stream>>>(xyz, nidx, stats);
    locse_prep <<<1, 32, 0, stream>>>(W, bias, gamma, beta, stats, Wp, bp);
    locse_main <<<4096, 256, 0, stream>>>(xyz, nidx, Wp, bp, out);
}